// MessageUpdatePore_44367012168459
// MI455X (gfx1250) — compile-verified
//
#include <hip/hip_runtime.h>
#include <math.h>

#define B_    2
#define E_    2048
#define N1_   96
#define K_    32
#define CIN_  64
#define CB_   32
#define COUT_ 64
#define G_    4
#define F_    (2*CIN_ + CB_)   /* 160 */
#define NSTEP (F_ / 4)         /* 40 WMMA k-steps */
#define NTILE (COUT_ / 16)     /* 4 N tiles */
#define FPAD  (F_ + 4)         /* LDS row stride: 164 dwords -> conflict-free */
#define NEG_SLOPE 0.01f

typedef float v2f __attribute__((ext_vector_type(2)));
typedef float v8f __attribute__((ext_vector_type(8)));

// ---------------------------------------------------------------------------
// Kernel 0: build B-matrix fragments in WMMA lane order and zero the output.
//   WbarP[(nt*NSTEP + s)*32 + lane] = { Wbar[4s+2*hi+0, o], Wbar[4s+2*hi+1, o] }
//   with Wbar[f,o] = mean_g W_eq[g,f,o], hi = lane>>4, o = nt*16 + (lane&15).
// One thread per fragment (v2f): 4*40*32 = 5120 threads.
// ---------------------------------------------------------------------------
__global__ void prep_kernel(const float* __restrict__ W_eq,
                            v2f* __restrict__ WbarP,
                            float* __restrict__ out) {
  int t = blockIdx.x * blockDim.x + threadIdx.x;
  if (t < NTILE * NSTEP * 32) {
    const int lane = t & 31;
    const int s    = (t >> 5) % NSTEP;
    const int nt   = t / (32 * NSTEP);
    const int hi   = lane >> 4;
    const int nl   = lane & 15;
    const int o    = nt * 16 + nl;
    v2f frag;
#pragma unroll
    for (int v = 0; v < 2; ++v) {
      const int f = 4 * s + 2 * hi + v;
      float sum = 0.f;
#pragma unroll
      for (int g = 0; g < G_; ++g) sum += W_eq[(g * F_ + f) * COUT_ + o];
      frag[v] = sum * (1.0f / (float)G_);
    }
    WbarP[t] = frag;
  }
  if (t < B_ * K_ * COUT_) out[t] = 0.f;
}

// ---------------------------------------------------------------------------
// Kernel 1: fused gather -> (4096x160)x(160x64) f32 WMMA GEMM -> bias ->
// leaky_relu -> sigmoid attention -> scatter-add into (B,K,COUT).
// 128 threads = 4 wave32; wave w owns output columns [16w, 16w+16).
// ---------------------------------------------------------------------------
__global__ __launch_bounds__(128)
void fused_kernel(const float* __restrict__ sites1,
                  const float* __restrict__ sites2,
                  const float* __restrict__ bonds,
                  const v2f*  __restrict__ WbarP,
                  const float* __restrict__ b_eq,
                  const float* __restrict__ W_att,
                  const float* __restrict__ b_att,
                  const int*   __restrict__ idx1,
                  const int*   __restrict__ idx2,
                  float* __restrict__ out) {
  __shared__ float vec[16][FPAD];  // 16 edge rows x 160 features (padded)
  __shared__ float attsum[16];
  __shared__ float attval[16];
  __shared__ int   ridx2[16];

  const int tid  = threadIdx.x;
  const int row0 = blockIdx.x * 16;

  // Gather the 16 feature rows: [sites1[idx1] | sites2[idx2] | bonds]
  for (int i = tid; i < 16 * F_; i += 128) {
    int r  = i / F_;
    int f  = i - r * F_;
    int gr = row0 + r;
    int b  = gr / E_;
    int e  = gr - b * E_;
    float v;
    if (f < CIN_) {
      v = sites1[(b * N1_ + idx1[e]) * CIN_ + f];
    } else if (f < 2 * CIN_) {
      v = sites2[(b * K_ + idx2[e]) * CIN_ + (f - CIN_)];
    } else {
      v = bonds[(b * E_ + e) * CB_ + (f - 2 * CIN_)];
    }
    vec[r][f] = v;
  }
  if (tid < 16) {
    attsum[tid] = 0.f;
    int gr = row0 + tid;
    int b  = gr / E_;
    int e  = gr - b * E_;
    ridx2[tid] = idx2[e];
  }
  __syncthreads();

  const int wave = tid >> 5;       // N-tile index (0..3)
  const int lane = tid & 31;
  const int nl   = lane & 15;      // A: M row / B,D: N col
  const int hi   = lane >> 4;      // K half-select for A/B, M+8 for D
  const int n0   = wave * 16;

  // Pre-swizzled B fragments: contiguous b64 loads at immediate offsets.
  const v2f* bptr = WbarP + (size_t)wave * NSTEP * 32 + lane;

  // D = A(16x4 f32) x B(4x16 f32) + C, stepped over F=160 in chunks of 4.
  v8f c = {};
#pragma unroll
  for (int s = 0; s < NSTEP; ++s) {
    const int kk = 4 * s + 2 * hi;
    v2f a;
    a.x = vec[nl][kk];
    a.y = vec[nl][kk + 1];
    const v2f bf = bptr[s * 32];
    c = __builtin_amdgcn_wmma_f32_16x16x4_f32(false, a, false, bf,
                                              (short)0, c, false, false);
  }

  const int   o   = n0 + nl;       // output channel
  const float be  = b_eq[o];
  const float wat = W_att[o];      // W_att[o, 0]

  float val[8];
#pragma unroll
  for (int r = 0; r < 8; ++r) {
    float v = c[r] + be;
    v = (v > 0.f) ? v : NEG_SLOPE * v;         // leaky_relu
    val[r] = v;
    // attention logit partial: sum_o lat[row,o] * W_att[o]  (row = r + 8*hi)
    atomicAdd(&attsum[r + 8 * hi], v * wat);   // ds_add_f32
  }
  __syncthreads();
  if (tid < 16) {
    float x = attsum[tid] + b_att[0];
    attval[tid] = 1.0f / (1.0f + __expf(-x));  // sigmoid
  }
  __syncthreads();

#pragma unroll
  for (int r = 0; r < 8; ++r) {
    const int   rr  = r + 8 * hi;              // row within tile
    const int   gr  = row0 + rr;
    const int   b   = gr / E_;
    const float lat = attval[rr] * val[r];
    atomicAdd(&out[(b * K_ + ridx2[rr]) * COUT_ + o], lat);
  }
}

// ---------------------------------------------------------------------------
extern "C" void kernel_launch(void* const* d_in, const int* in_sizes, int n_in,
                              void* d_out, int out_size, void* d_ws, size_t ws_size,
                              hipStream_t stream) {
  // setup_inputs order:
  // 0 sites1, 1 sites2, 2 bonds, 3 W_eq, 4 b_eq, 5 W_att, 6 b_att,
  // 7 idx2_oh (unused), 8 idx1, 9 idx2, 10 perms1 (unused), 11 perms2 (unused)
  const float* sites1 = (const float*)d_in[0];
  const float* sites2 = (const float*)d_in[1];
  const float* bonds  = (const float*)d_in[2];
  const float* W_eq   = (const float*)d_in[3];
  const float* b_eq   = (const float*)d_in[4];
  const float* W_att  = (const float*)d_in[5];
  const float* b_att  = (const float*)d_in[6];
  const int*   idx1   = (const int*)d_in[8];
  const int*   idx2   = (const int*)d_in[9];

  float* out   = (float*)d_out;
  v2f*   WbarP = (v2f*)d_ws;   // NTILE*NSTEP*32 v2f = 40 KB fragment buffer

  const int prep_threads = NTILE * NSTEP * 32;  // 5120 (covers 4096 out zeroing)
  prep_kernel<<<(prep_threads + 255) / 256, 256, 0, stream>>>(W_eq, WbarP, out);

  fused_kernel<<<(B_ * E_) / 16, 128, 0, stream>>>(
      sites1, sites2, bonds, WbarP, b_eq, W_att, b_att, idx1, idx2, out);
}